// CTCLayer_35888746725989
// MI455X (gfx1250) — compile-verified
//
#include <hip/hip_runtime.h>
#include <math.h>

// ---- problem constants (match reference) ----
#define BB 512
#define TT 512
#define CC 96      // classes, blank = CC-1
#define LL 100     // max label length
#define SS 201     // 2*LL + 1 extended states
#define EPSF 1e-7f
#define NEGF (-1e30f)

typedef __attribute__((ext_vector_type(2))) float v2f;
typedef __attribute__((ext_vector_type(8))) float v8f;

// -------------------------------------------------------------------------
// Kernel 1: denom[b*T + t] = log( sum_c (y[b,t,c]) + C*eps )
// One wave32 handles 16 rows; row-sum over C=96 done as 24 chained
// V_WMMA_F32_16X16X4_F32 ops against an all-ones B matrix.
// A layout (32-bit 16x4): lanes 0-15 hold M=lane, VGPR0=K0,VGPR1=K1;
// lanes 16-31 hold M=lane-16, VGPR0=K2,VGPR1=K3.
// D layout: VGPR r -> M=r (lanes 0-15) / M=8+r (lanes 16-31), all N equal.
// -------------------------------------------------------------------------
__global__ __launch_bounds__(256) void ctc_denom_wmma(const float* __restrict__ y,
                                                      float* __restrict__ denom,
                                                      int nrows) {
    const int lane = threadIdx.x & 31;
    const int gwave = (blockIdx.x * blockDim.x + threadIdx.x) >> 5;
    const int r0 = gwave * 16;
    if (r0 >= nrows) return;                  // uniform per wave -> EXEC all-1s inside

    const int half = lane >> 4;               // 0: K+0/K+1, 1: K+2/K+3
    const int row  = r0 + (lane & 15);
    const float* p = y + (size_t)row * CC + 2 * half;

    v8f acc = {};
    v2f bones; bones.x = 1.0f; bones.y = 1.0f;
#pragma unroll
    for (int c = 0; c < 24; ++c) {            // 24 * 4 = 96 = C
        v2f a;
        a.x = p[c * 4 + 0];
        a.y = p[c * 4 + 1];
        acc = __builtin_amdgcn_wmma_f32_16x16x4_f32(
            /*neg_a=*/false, a, /*neg_b=*/false, bones,
            /*c_mod=*/(short)0, acc, /*reuse_a=*/false, /*reuse_b=*/false);
    }
    const float ce = (float)CC * EPSF;
    if (lane == 0) {
#pragma unroll
        for (int r = 0; r < 8; ++r) denom[r0 + r] = logf(acc[r] + ce);
    } else if (lane == 16) {
#pragma unroll
        for (int r = 0; r < 8; ++r) denom[r0 + 8 + r] = logf(acc[r] + ce);
    }
}

// -------------------------------------------------------------------------
// Kernel 2: CTC forward recursion. One block (256 threads, 8 waves) per
// batch element; thread s owns extended state s (s < 201). Alpha kept in
// double-buffered LDS; y_pred row double-buffered in LDS so one barrier
// per timestep suffices; next row prefetched with global_prefetch.
// -------------------------------------------------------------------------
__global__ __launch_bounds__(256) void ctc_forward(const float* __restrict__ y,
                                                   const int*   __restrict__ labels,
                                                   const int*   __restrict__ inlen_arr,
                                                   const int*   __restrict__ lablen_arr,
                                                   const float* __restrict__ denom, // may be null
                                                   float* __restrict__ out) {
    const int b   = blockIdx.x;
    const int tid = threadIdx.x;

    __shared__ float bufA[SS + 7];
    __shared__ float bufB[SS + 7];
    __shared__ float afin[SS + 7];
    __shared__ float rowBuf[2][CC];

    const int blank  = CC - 1;
    int inlen  = inlen_arr[b];
    if (inlen < 1)  inlen = 1;
    if (inlen > TT) inlen = TT;
    const int lablen = lablen_arr[b];

    // per-thread static state for extended label position s = tid
    const int  s      = tid;
    const bool active = (s < SS);
    int  my_ext   = blank;
    bool my_skip  = false;
    bool my_valid = false;
    if (active) {
        my_valid = (s < 2 * lablen + 1);
        if (s & 1) {
            const int li = (s - 1) >> 1;
            my_ext = labels[b * LL + li];
            if (s >= 3) my_skip = (labels[b * LL + li] != labels[b * LL + li - 1]);
        }
    }

    const float* yb = y + (size_t)b * TT * CC;
    const bool have_denom = (denom != nullptr);
    const float* db = have_denom ? (denom + (size_t)b * TT) : nullptr;

    // ---- t = 0 ----
    if (tid < CC) rowBuf[0][tid] = yb[tid];
    __syncthreads();

    float dd;
    if (have_denom) {
        dd = db[0];
    } else {
        const int lane = tid & 31;
        float ssum = rowBuf[0][lane] + rowBuf[0][lane + 32] + rowBuf[0][lane + 64];
        ssum += __shfl_xor(ssum, 16, 32);
        ssum += __shfl_xor(ssum, 8, 32);
        ssum += __shfl_xor(ssum, 4, 32);
        ssum += __shfl_xor(ssum, 2, 32);
        ssum += __shfl_xor(ssum, 1, 32);
        dd = logf(ssum + (float)CC * EPSF);
    }

    float* cur = bufA;
    float* nxt = bufB;
    if (active) {
        const float lp0 = logf(rowBuf[0][my_ext] + EPSF) - dd;
        float a0 = (s <= 1) ? lp0 : NEGF;
        if (!my_valid) a0 = NEGF;
        cur[s] = a0;
        if (inlen == 1) afin[s] = a0;
    }

    // ---- t = 1 .. T-1 : one __syncthreads per step ----
    for (int t = 1; t < TT; ++t) {
        if (tid < CC) {
            rowBuf[t & 1][tid] = yb[(size_t)t * CC + tid];
            if (t + 1 < TT)
                __builtin_prefetch(&yb[(size_t)(t + 1) * CC + tid], 0, 1);
        }
        __syncthreads();   // row(t) ready; cur fully written by previous step

        if (have_denom) {
            dd = db[t];
        } else {
            const int lane = tid & 31;
            float ssum = rowBuf[t & 1][lane] + rowBuf[t & 1][lane + 32] +
                         rowBuf[t & 1][lane + 64];
            ssum += __shfl_xor(ssum, 16, 32);
            ssum += __shfl_xor(ssum, 8, 32);
            ssum += __shfl_xor(ssum, 4, 32);
            ssum += __shfl_xor(ssum, 2, 32);
            ssum += __shfl_xor(ssum, 1, 32);
            dd = logf(ssum + (float)CC * EPSF);
        }

        if (active) {
            const float lp = logf(rowBuf[t & 1][my_ext] + EPSF) - dd;
            const float a1 = cur[s];
            const float a2 = (s >= 1) ? cur[s - 1] : NEGF;
            const float a3 = my_skip ? cur[s - 2] : NEGF;
            const float m  = fmaxf(a1, fmaxf(a2, a3));
            const float lse = m + logf(expf(a1 - m) + expf(a2 - m) + expf(a3 - m));
            const float val = my_valid ? (lse + lp) : NEGF;
            nxt[s] = val;
            if (t == inlen - 1) afin[s] = val;
        }
        float* tmp = cur; cur = nxt; nxt = tmp;
    }

    __syncthreads();
    if (tid == 0) {
        const int last = 2 * lablen;             // in [2, 200] <= SS-1
        const float l1 = afin[last];
        const float l2 = afin[last - 1];
        const float m  = fmaxf(l1, l2);
        out[b] = -(m + logf(expf(l1 - m) + expf(l2 - m)));
    }
}

// -------------------------------------------------------------------------
// Launch: inputs in setup_inputs() order:
//   d_in[0] y_true  [B,L]  int32
//   d_in[1] y_pred  [B,T,C] float32
//   d_in[2] input_length [B] int32
//   d_in[3] label_length [B] int32
// d_out: [B] float32
// -------------------------------------------------------------------------
extern "C" void kernel_launch(void* const* d_in, const int* in_sizes, int n_in,
                              void* d_out, int out_size, void* d_ws, size_t ws_size,
                              hipStream_t stream) {
    const int*   y_true  = (const int*)d_in[0];
    const float* y_pred  = (const float*)d_in[1];
    const int*   in_len  = (const int*)d_in[2];
    const int*   lab_len = (const int*)d_in[3];
    float*       out     = (float*)d_out;

    const int Bv    = in_sizes[2];      // batch
    const int nrows = Bv * TT;          // B*T rows of length C

    float* denom = nullptr;
    if (ws_size >= (size_t)nrows * sizeof(float)) {
        denom = (float*)d_ws;
        // 16 rows per wave, 8 waves per 256-thread block
        const int waves  = nrows / 16;
        const int blocks = (waves + 7) / 8;
        ctc_denom_wmma<<<blocks, 256, 0, stream>>>(y_pred, denom, nrows);
    }

    ctc_forward<<<Bv, 256, 0, stream>>>(y_pred, y_true, in_len, lab_len, denom, out);
}